// GCN_6605659701468
// MI455X (gfx1250) — compile-verified
//
#include <hip/hip_runtime.h>

typedef __attribute__((ext_vector_type(2))) float v2f;
typedef __attribute__((ext_vector_type(8))) float v8f;

#define FOUT 16

__device__ __forceinline__ void atomAddF(float* p, float v) {
#if defined(__HIP_DEVICE_COMPILE__)
    unsafeAtomicAdd(p, v);   // fire-and-forget global_atomic_add_f32
#else
    (void)p; (void)v;
#endif
}

// ---------------------------------------------------------------------------
// GEMM: Y[N,16] = X[N,K] @ W[K,16] via V_WMMA_F32_16X16X4_F32.
// One wave per 16-row tile. K must be a multiple of 4 (512 and 16 here).
// A layout (16x4 f32): lanes 0-15 hold K={0,1} in v0/v1 for row M=lane;
//                      lanes 16-31 hold K={2,3} for row M=lane-16.
// B layout (4x16 f32): lanes 0-15 hold rows K={0,1} at col N=lane;
//                      lanes 16-31 hold rows K={2,3} at col N=lane-16.
// C/D (16x16 f32): VGPR r -> M=r (lanes 0-15) / M=r+8 (lanes 16-31), N=lane&15.
// ---------------------------------------------------------------------------
__global__ __launch_bounds__(128) void gemm_wmma_f32(
    const float* __restrict__ X, const float* __restrict__ W,
    float* __restrict__ Y, int N, int K)
{
    int wave = (int)((blockIdx.x * blockDim.x + threadIdx.x) >> 5);
    int lane = (int)(threadIdx.x & 31);
    int row0 = wave * 16;
    if (row0 >= N) return;            // wave-uniform: EXEC stays all-ones below

    int m  = lane & 15;
    int kh = lane >> 4;               // 0 -> K pair {0,1}; 1 -> K pair {2,3}
    const float* arow = X + (size_t)(row0 + m) * (size_t)K;

    v8f acc = {0.f, 0.f, 0.f, 0.f, 0.f, 0.f, 0.f, 0.f};
    for (int k = 0; k < K; k += 4) {
        const float* ap = arow + k + kh * 2;
        v2f A; A.x = ap[0]; A.y = ap[1];
        const float* bp = W + (size_t)(k + kh * 2) * FOUT + m;
        v2f B; B.x = bp[0]; B.y = bp[FOUT];
        acc = __builtin_amdgcn_wmma_f32_16x16x4_f32(
            /*neg_a=*/false, A, /*neg_b=*/false, B,
            /*c_mod=*/(short)0, acc, /*reuse_a=*/false, /*reuse_b=*/false);
    }

    int mbase = row0 + (kh << 3);
    #pragma unroll
    for (int r = 0; r < 8; ++r)
        Y[(size_t)(mbase + r) * FOUT + m] = acc[r];
}

// ---------------------------------------------------------------------------
__global__ void deg_init(float* __restrict__ deg, int n) {
    int i = blockIdx.x * blockDim.x + threadIdx.x;
    if (i < n) deg[i] = 1.0f;                       // self-loop contribution
}

__global__ void deg_acc(const int* __restrict__ dst, float* __restrict__ deg,
                        long long E) {
    long long e = (long long)blockIdx.x * blockDim.x + threadIdx.x;
    if (e < E) atomAddF(&deg[dst[e]], 1.0f);
}

__global__ void deg_to_dinv(float* __restrict__ deg, int n) {
    int i = blockIdx.x * blockDim.x + threadIdx.x;
    if (i < n) deg[i] = __frsqrt_rn(deg[i]);        // deg >= 1 always
}

__global__ void zero_f32(float* __restrict__ p, long long n) {
    long long i = (long long)blockIdx.x * blockDim.x + threadIdx.x;
    if (i < n) p[i] = 0.0f;
}

// 16 threads per edge: coalesced 64B gather of xw[src][:] and 64B of atomics
// into agg[dst][:]. xw/agg are ~6.4MB each -> L2 resident.
__global__ void edge_agg(const int* __restrict__ src, const int* __restrict__ dst,
                         const float* __restrict__ dinv,
                         const float* __restrict__ xw, float* __restrict__ agg,
                         long long E)
{
    long long gid = (long long)blockIdx.x * blockDim.x + threadIdx.x;
    long long e = gid >> 4;
    int j = (int)(gid & 15);
    if (e >= E) return;
    int s = src[e];
    int d = dst[e];
    float coeff = dinv[s] * dinv[d];
    float v = xw[(size_t)s * FOUT + j] * coeff;
    atomAddF(&agg[(size_t)d * FOUT + j], v);
}

// h = relu(agg + dinv^2 * xw + b)   (self-loop message fused here)
__global__ void finish_layer(const float* __restrict__ agg,
                             const float* __restrict__ xw,
                             const float* __restrict__ dinv,
                             const float* __restrict__ b,
                             float* __restrict__ h, int n)
{
    long long gid = (long long)blockIdx.x * blockDim.x + threadIdx.x;
    int i = (int)(gid >> 4);
    int j = (int)(gid & 15);
    if (i >= n) return;
    float di = dinv[i];
    size_t idx = (size_t)i * FOUT + j;
    float v = agg[idx] + di * di * xw[idx] + b[j];
    h[idx] = fmaxf(v, 0.0f);
}

// Column sums of h[N,16] -> sums[16]; grid-strided partials, one atomic/thread.
__global__ void pool_sum(const float* __restrict__ h, float* __restrict__ sums,
                         int n)
{
    int tid = blockIdx.x * blockDim.x + threadIdx.x;
    int j = tid & 15;
    int slot = tid >> 4;
    int nslots = (gridDim.x * blockDim.x) >> 4;
    float acc = 0.0f;
    for (int i = slot; i < n; i += nslots)
        acc += h[(size_t)i * FOUT + j];
    atomAddF(&sums[j], acc);
}

__global__ void final_fc(const float* __restrict__ sums,
                         const float* __restrict__ fc_w,
                         const float* __restrict__ fc_b,
                         float* __restrict__ out, float invN)
{
    if (blockIdx.x == 0 && threadIdx.x == 0) {
        float acc = fc_b[0];
        #pragma unroll
        for (int j = 0; j < FOUT; ++j)
            acc += sums[j] * invN * fc_w[j];
        out[0] = acc;
    }
}

// ---------------------------------------------------------------------------
extern "C" void kernel_launch(void* const* d_in, const int* in_sizes, int n_in,
                              void* d_out, int out_size, void* d_ws, size_t ws_size,
                              hipStream_t stream)
{
    const float* x    = (const float*)d_in[0];
    const int*   ei   = (const int*)d_in[1];
    const float* W1   = (const float*)d_in[2];
    const float* b1   = (const float*)d_in[3];
    const float* W2   = (const float*)d_in[4];
    const float* b2   = (const float*)d_in[5];
    const float* fc_w = (const float*)d_in[6];
    const float* fc_b = (const float*)d_in[7];

    const int K1 = in_sizes[2] / FOUT;            // 512
    const int N  = in_sizes[0] / K1;              // 100000
    const long long E = in_sizes[1] / 2;          // 3,200,000
    const int* src = ei;
    const int* dst = ei + E;

    // Workspace layout (floats): xw[N*16] | h[N*16] | agg[N*16] | dinv[N] | sums[16]
    float* f_xw  = (float*)d_ws;
    float* f_h   = f_xw + (size_t)N * FOUT;
    float* f_agg = f_h  + (size_t)N * FOUT;
    float* f_di  = f_agg + (size_t)N * FOUT;
    float* f_sum = f_di + (size_t)N;

    const int T = 256;
    const long long Nf      = (long long)N * FOUT;
    const int  blkN         = (N + T - 1) / T;
    const int  blkNf        = (int)((Nf + T - 1) / T);
    const int  blkE         = (int)((E + T - 1) / T);
    const int  blkE16       = (int)((E * FOUT + T - 1) / T);
    const int  nwaves       = (N + 15) / 16;
    const int  blkGemm      = (nwaves * 32 + 127) / 128;

    // --- shared degree normalization (identical for both layers) ---
    deg_init<<<blkN, T, 0, stream>>>(f_di, N);
    deg_acc<<<blkE, T, 0, stream>>>(dst, f_di, E);
    deg_to_dinv<<<blkN, T, 0, stream>>>(f_di, N);

    // --- layer 1 ---
    gemm_wmma_f32<<<blkGemm, 128, 0, stream>>>(x, W1, f_xw, N, K1);
    zero_f32<<<blkNf, T, 0, stream>>>(f_agg, Nf);
    edge_agg<<<blkE16, T, 0, stream>>>(src, dst, f_di, f_xw, f_agg, E);
    finish_layer<<<blkNf, T, 0, stream>>>(f_agg, f_xw, f_di, b1, f_h, N);

    // --- layer 2 ---
    gemm_wmma_f32<<<blkGemm, 128, 0, stream>>>(f_h, W2, f_xw, N, FOUT);
    zero_f32<<<blkNf, T, 0, stream>>>(f_agg, Nf);
    edge_agg<<<blkE16, T, 0, stream>>>(src, dst, f_di, f_xw, f_agg, E);
    finish_layer<<<blkNf, T, 0, stream>>>(f_agg, f_xw, f_di, b2, f_h, N);

    // --- global mean pool + FC ---
    zero_f32<<<1, 32, 0, stream>>>(f_sum, 16);
    pool_sum<<<256, T, 0, stream>>>(f_h, f_sum, N);
    final_fc<<<1, 32, 0, stream>>>(f_sum, fc_w, fc_b, (float*)d_out, 1.0f / (float)N);
}